// QuantumBiasedAttention_62380105007997
// MI455X (gfx1250) — compile-verified
//
#include <hip/hip_runtime.h>
#include <hip/hip_bf16.h>

#define NORB 96
#define CH   128
#define NH   16
#define HD   8

typedef __attribute__((ext_vector_type(2))) float v2f;
typedef __attribute__((ext_vector_type(8))) float v8f;

// ---------------------------------------------------------------------------
// Kernel 1: fused J / K contraction, streaming eri exactly once (340 MB).
// One block per (a,b) slab of 96x96 contiguous floats. Lane t owns column d=t.
//   J[a,b]      = sum_{c,d} P[c,d] * slab[c,d]
//   Kpart[a,b,d]= sum_{c}   P[a,c] * slab[c,d]   (later reduced over a)
// eri uses non-temporal loads so P stays resident in WGP$/L2.
// ---------------------------------------------------------------------------
__global__ void fused_jk_kernel(const float* __restrict__ eri,
                                const float* __restrict__ P,
                                float* __restrict__ J,
                                float* __restrict__ Kpart) {
  const int ab = blockIdx.x;            // 0..9215, ab = a*96 + b
  const int a  = ab / NORB;
  const int t  = threadIdx.x;           // 0..95 (3 waves)
  const float* __restrict__ slab = eri + (size_t)ab * (NORB * NORB);
  const float* __restrict__ Pa   = P + a * NORB;

  float jacc = 0.f, kacc = 0.f;
  #pragma unroll 1
  for (int c = 0; c < NORB; c += 4) {
    if (c + 16 < NORB) {
      __builtin_prefetch(slab + (c + 16) * NORB + t, 0, 0);   // global_prefetch
    }
    float v0 = __builtin_nontemporal_load(slab + (c + 0) * NORB + t);
    float v1 = __builtin_nontemporal_load(slab + (c + 1) * NORB + t);
    float v2 = __builtin_nontemporal_load(slab + (c + 2) * NORB + t);
    float v3 = __builtin_nontemporal_load(slab + (c + 3) * NORB + t);
    jacc = fmaf(P[(c + 0) * NORB + t], v0, jacc);
    jacc = fmaf(P[(c + 1) * NORB + t], v1, jacc);
    jacc = fmaf(P[(c + 2) * NORB + t], v2, jacc);
    jacc = fmaf(P[(c + 3) * NORB + t], v3, jacc);
    kacc = fmaf(Pa[c + 0], v0, kacc);
    kacc = fmaf(Pa[c + 1], v1, kacc);
    kacc = fmaf(Pa[c + 2], v2, kacc);
    kacc = fmaf(Pa[c + 3], v3, kacc);
  }
  // partial K row for this (a,b): deterministic (no float atomics)
  Kpart[(size_t)ab * NORB + t] = kacc;

  // reduce jacc across 96 threads -> J[ab]
  __shared__ float red[NORB];
  red[t] = jacc;
  __syncthreads();
  if (t < 32) {
    float s = red[t] + red[t + 32] + red[t + 64];
    #pragma unroll
    for (int off = 16; off > 0; off >>= 1) s += __shfl_down(s, off, 32);
    if (t == 0) J[ab] = s;
  }
}

// K[b,d] = sum_a Kpart[(a*96+b)*96 + d]
__global__ void k_reduce_kernel(const float* __restrict__ Kpart,
                                float* __restrict__ Kout) {
  const int b = blockIdx.x;
  const int d = threadIdx.x;  // 0..95
  float s = 0.f;
  #pragma unroll 4
  for (int a = 0; a < NORB; ++a) {
    s += __builtin_nontemporal_load(Kpart + ((size_t)(a * NORB + b)) * NORB + d);
  }
  Kout[b * NORB + d] = s;
}

// ---------------------------------------------------------------------------
// f32 WMMA GEMM: C[M,N] = A[M,K] * op(B) (+ bias). K/TRANSB/HASBIAS are
// compile-time so the inner loop is a branch-free, fully unrolled stream of
// global_load_b64 + v_wmma_f32_16x16x4_f32. One wave per 16x16 C tile.
// A frag layout: lane = m + 16*(k/2), vgpr = k%2 ;  B frag: lane = n + 16*(k/2).
// C layout: lanes 0-15 vgpr r -> (m0+r, n0+lane); lanes 16-31 -> (m0+r+8, ...).
// ---------------------------------------------------------------------------
template <int K, bool TRANSB, bool HASBIAS>
__global__ void wmma_gemm_f32(const float* __restrict__ A,
                              const float* __restrict__ B,
                              const float* __restrict__ bias,
                              float* __restrict__ C,
                              int lda, int ldb, int ldc) {
  const int n0   = blockIdx.x * 16;
  const int m0   = blockIdx.y * 16;
  const int lane = threadIdx.x;     // 0..31, EXEC all ones
  const int half = lane >> 4;
  const int lm   = lane & 15;

  const float* __restrict__ Arow = A + (m0 + lm) * lda;
  const float* __restrict__ Bcol = TRANSB ? (B + (n0 + lm) * ldb) : (B + n0 + lm);

  v8f acc = {};
  #pragma unroll
  for (int kk = 0; kk < K; kk += 4) {
    const int ka = kk + 2 * half;
    v2f a, b;
    a.x = Arow[ka];
    a.y = Arow[ka + 1];
    if (TRANSB) {               // B[k,n] = Bt[n,k]
      b.x = Bcol[ka];
      b.y = Bcol[ka + 1];
    } else {
      b.x = Bcol[(ka + 0) * ldb];
      b.y = Bcol[(ka + 1) * ldb];
    }
    acc = __builtin_amdgcn_wmma_f32_16x16x4_f32(false, a, false, b,
                                                (short)0, acc, false, false);
  }
  #pragma unroll
  for (int r = 0; r < 8; ++r) {
    const int m = m0 + r + 8 * half;
    const int n = n0 + lm;
    float v = acc[r];
    if (HASBIAS) v += bias[n];
    C[m * ldc + n] = v;
  }
}

// ---------------------------------------------------------------------------
// LayerNorm over feature dim (C=128), one block per row.
// ---------------------------------------------------------------------------
__global__ void layernorm_kernel(const float* __restrict__ x,
                                 const float* __restrict__ scale,
                                 const float* __restrict__ bias,
                                 float* __restrict__ xn) {
  const int i = blockIdx.x;
  const int t = threadIdx.x;  // 0..127
  __shared__ float red[CH];
  const float v = x[i * CH + t];

  red[t] = v; __syncthreads();
  for (int off = 64; off > 0; off >>= 1) { if (t < off) red[t] += red[t + off]; __syncthreads(); }
  const float mu = red[0] * (1.0f / CH);
  __syncthreads();

  const float d = v - mu;
  red[t] = d * d; __syncthreads();
  for (int off = 64; off > 0; off >>= 1) { if (t < off) red[t] += red[t + off]; __syncthreads(); }
  const float var = red[0] * (1.0f / CH);

  const float r = rsqrtf(var + 1e-6f);
  xn[i * CH + t] = d * r * scale[t] + bias[t];
}

// ---------------------------------------------------------------------------
// Attention with per-head quantum bias. One block per (head, row i).
// qkv layout: [N, 3C]; q at col h*8, k at 128+h*8, v at 256+h*8.
// ---------------------------------------------------------------------------
__global__ void attn_kernel(const float* __restrict__ qkv,
                            const float* __restrict__ Hc,
                            const float* __restrict__ J,
                            const float* __restrict__ Km,
                            const float* __restrict__ L,
                            float* __restrict__ att) {
  const int h = blockIdx.x;
  const int i = blockIdx.y;
  const int j = threadIdx.x;  // 0..95
  __shared__ float qi[HD];
  __shared__ float red[NORB];
  __shared__ float prob[NORB];
  __shared__ float mx_s, sum_s;

  if (j < HD) qi[j] = qkv[i * (3 * CH) + h * HD + j];
  __syncthreads();

  float s = 0.f;
  const float* __restrict__ krow = qkv + j * (3 * CH) + CH + h * HD;
  #pragma unroll
  for (int d = 0; d < HD; ++d) s = fmaf(qi[d], krow[d], s);
  s *= 0.35355339059327373f;  // 1/sqrt(8)

  if      (h < 2) s += Hc[i * NORB + j];
  else if (h < 4) s += J[i * NORB + j] - 0.5f * Km[i * NORB + j];
  else if (h < 6) s += L[i * NORB + j];

  // softmax over j (96 lanes, 3 waves)
  red[j] = s; __syncthreads();
  if (j < 32) {
    float m = fmaxf(red[j], fmaxf(red[j + 32], red[j + 64]));
    #pragma unroll
    for (int off = 16; off > 0; off >>= 1) m = fmaxf(m, __shfl_down(m, off, 32));
    if (j == 0) mx_s = m;
  }
  __syncthreads();
  const float p = __expf(s - mx_s);
  red[j] = p; __syncthreads();
  if (j < 32) {
    float m = red[j] + red[j + 32] + red[j + 64];
    #pragma unroll
    for (int off = 16; off > 0; off >>= 1) m += __shfl_down(m, off, 32);
    if (j == 0) sum_s = m;
  }
  __syncthreads();
  prob[j] = p / sum_s;
  __syncthreads();

  if (j < HD) {
    float acc = 0.f;
    const float* __restrict__ vcol = qkv + 2 * CH + h * HD + j;
    #pragma unroll 4
    for (int jj = 0; jj < NORB; ++jj) acc = fmaf(prob[jj], vcol[jj * (3 * CH)], acc);
    att[i * CH + h * HD + j] = acc;
  }
}

// ---------------------------------------------------------------------------
// Host-side orchestration (graph-capture safe: only kernel launches).
// ---------------------------------------------------------------------------
extern "C" void kernel_launch(void* const* d_in, const int* in_sizes, int n_in,
                              void* d_out, int out_size, void* d_ws, size_t ws_size,
                              hipStream_t stream) {
  const float* x    = (const float*)d_in[0];
  const float* Hc   = (const float*)d_in[1];
  const float* X    = (const float*)d_in[2];
  const float* eri  = (const float*)d_in[3];
  const float* P    = (const float*)d_in[4];
  const float* lnsc = (const float*)d_in[5];
  const float* lnb  = (const float*)d_in[6];
  const float* Wqkv = (const float*)d_in[7];
  const float* Wout = (const float*)d_in[8];
  const float* bout = (const float*)d_in[9];
  float* out = (float*)d_out;

  float* ws    = (float*)d_ws;
  float* Kpart = ws;                       // 96*96*96 = 884736
  float* Jw    = Kpart + 884736;           // 9216
  float* Kw    = Jw + 9216;                // 9216
  float* Lw    = Kw + 9216;                // 9216
  float* Tw    = Lw + 9216;                // 9216
  float* xn    = Tw + 9216;                // 12288
  float* qkv   = xn + 12288;               // 36864
  float* att   = qkv + 36864;              // 12288

  // 1) fused J/K: single streaming pass over 340MB eri (the whole runtime)
  fused_jk_kernel<<<NORB * NORB, NORB, 0, stream>>>(eri, P, Jw, Kpart);
  // 2) deterministic K reduction over a
  k_reduce_kernel<<<NORB, NORB, 0, stream>>>(Kpart, Kw);
  // 3) T = X @ H_core ; L = T @ X^T   (WMMA f32)
  wmma_gemm_f32<96, false, false><<<dim3(6, 6), 32, 0, stream>>>(X,  Hc, nullptr, Tw, 96, 96, 96);
  wmma_gemm_f32<96, true,  false><<<dim3(6, 6), 32, 0, stream>>>(Tw, X,  nullptr, Lw, 96, 96, 96);
  // 4) LayerNorm
  layernorm_kernel<<<NORB, CH, 0, stream>>>(x, lnsc, lnb, xn);
  // 5) qkv = xn @ Wqkv   (96x384x128, WMMA f32)
  wmma_gemm_f32<128, false, false><<<dim3(24, 6), 32, 0, stream>>>(xn, Wqkv, nullptr, qkv, 128, 384, 384);
  // 6) biased multi-head attention -> att [96,128]
  attn_kernel<<<dim3(NH, NORB), NORB, 0, stream>>>(qkv, Hc, Jw, Kw, Lw, att);
  // 7) out = att @ Wout + bout   (96x128x128, WMMA f32)
  wmma_gemm_f32<128, false, true><<<dim3(8, 6), 32, 0, stream>>>(att, Wout, bout, out, 128, 128, 128);
}